// OHEM_46497315946561
// MI455X (gfx1250) — compile-verified
//
#include <hip/hip_runtime.h>

typedef __attribute__((ext_vector_type(2))) float v2f;
typedef __attribute__((ext_vector_type(8))) float v8f;

#define LN2F 0.69314718055994530942f
#define LOG_CLAMP -100.0f
#define HW_ELEMS (512 * 512)          // elements per sample
#define NSAMPLES 64
#define CHUNKS_PER_SAMPLE 32          // blocks per sample
#define NPARTIAL (NSAMPLES * CHUNKS_PER_SAMPLE)  // 2048
#define TOPK 42

// Torch-style clamped BCE element: -(t*log(p) + (1-t)*log1p(-p)), logs clamped at -100.
// v_log_f32 is log2; multiply by ln2. log2(0)=-inf -> clamp handles p==0 and p==1.
__device__ __forceinline__ float bce_elem(float p, float t) {
    float lp  = fmaxf(__log2f(p) * LN2F, LOG_CLAMP);
    float l1p = fmaxf(__log2f(1.0f - p) * LN2F, LOG_CLAMP);
    // -(t*lp + (1-t)*l1p) == -(l1p + t*(lp - l1p))
    return -fmaf(t, lp - l1p, l1p);
}

// Wave32 sum-reduction via V_WMMA_F32_16X16X4_F32 with B = ones.
// A layout (16x4 f32): lanes 0-15 hold K=0 (vgpr0), K=1 (vgpr1); lanes 16-31 hold K=2,K=3.
// Put lane value in vgpr0, zero in vgpr1 -> A[m][k] nonzero at (m=lane%16, k=2*(lane/16)).
// With B all-ones: D[m][n] = sum_k A[m][k] = val(lane m) + val(lane m+16), replicated over n.
// C/D layout: lanes 0-15 vgpr r = D[r][lane]; lanes 16-31 vgpr r = D[8+r][lane-16].
// Sum of a lane's 8 D vgprs = half-wave sum; xor-16 shuffle add completes the wave sum.
__device__ __forceinline__ float wave_reduce_sum_wmma(float v) {
    v2f a; a[0] = v;    a[1] = 0.0f;
    v2f b; b[0] = 1.0f; b[1] = 1.0f;
    v8f c = {};
    c = __builtin_amdgcn_wmma_f32_16x16x4_f32(
        /*neg_a=*/false, a, /*neg_b=*/false, b,
        /*c_mod=*/(short)0, c, /*reuse_a=*/false, /*reuse_b=*/false);
    float s = ((c[0] + c[1]) + (c[2] + c[3])) + ((c[4] + c[5]) + (c[6] + c[7]));
    s += __shfl_xor(s, 16, 32);
    return s;
}

// Stage 1: per-block partial BCE sums. 32 blocks per sample, 256 threads per block,
// each thread processes 8 float4's (32 elements) from each input: 8192 floats/block.
__global__ __launch_bounds__(256) void ohem_partial_kernel(
        const float* __restrict__ preds,
        const float* __restrict__ targets,
        float* __restrict__ partial) {
    const int sample = blockIdx.x >> 5;           // / CHUNKS_PER_SAMPLE
    const int chunk  = blockIdx.x & 31;
    const int base   = sample * HW_ELEMS + chunk * (HW_ELEMS / CHUNKS_PER_SAMPLE);

    const float4* __restrict__ p4 = (const float4*)(preds + base);
    const float4* __restrict__ t4 = (const float4*)(targets + base);

    const int tid = threadIdx.x;
    float sum = 0.0f;
#pragma unroll
    for (int i = 0; i < 8; ++i) {
        float4 p = p4[tid + i * 256];
        float4 t = t4[tid + i * 256];
        sum += bce_elem(p.x, t.x);
        sum += bce_elem(p.y, t.y);
        sum += bce_elem(p.z, t.z);
        sum += bce_elem(p.w, t.w);
    }

    // Wave-level reduction using WMMA (EXEC is all ones here: uniform control flow,
    // full 256-thread blocks).
    float wsum = wave_reduce_sum_wmma(sum);

    __shared__ float lds[8];
    const int lane = tid & 31;
    const int wave = tid >> 5;
    if (lane == 0) lds[wave] = wsum;
    __syncthreads();
    if (tid == 0) {
        float s = 0.0f;
#pragma unroll
        for (int i = 0; i < 8; ++i) s += lds[i];
        partial[blockIdx.x] = s;
    }
}

// Stage 2: fold partials -> per-sample mean loss -> mean of top-42.
__global__ __launch_bounds__(64) void ohem_final_kernel(
        const float* __restrict__ partial,
        float* __restrict__ out) {
    __shared__ float losses[NSAMPLES];
    __shared__ float contrib[NSAMPLES];
    const int i = threadIdx.x;

    float s = 0.0f;
    for (int j = 0; j < CHUNKS_PER_SAMPLE; ++j) s += partial[i * CHUNKS_PER_SAMPLE + j];
    const float loss = s * (1.0f / (float)HW_ELEMS);
    losses[i] = loss;
    __syncthreads();

    // Rank with top_k tie semantics (earlier index wins among equals).
    int rank = 0;
    for (int j = 0; j < NSAMPLES; ++j) {
        float lj = losses[j];
        rank += (lj > loss) || (lj == loss && j < i);
    }
    contrib[i] = (rank < TOPK) ? loss : 0.0f;
    __syncthreads();

    if (i == 0) {
        float acc = 0.0f;
        for (int j = 0; j < NSAMPLES; ++j) acc += contrib[j];
        out[0] = acc * (1.0f / (float)TOPK);
    }
}

extern "C" void kernel_launch(void* const* d_in, const int* in_sizes, int n_in,
                              void* d_out, int out_size, void* d_ws, size_t ws_size,
                              hipStream_t stream) {
    const float* preds   = (const float*)d_in[0];
    const float* targets = (const float*)d_in[1];
    float* partial = (float*)d_ws;   // NPARTIAL floats; fully overwritten each launch
    float* out = (float*)d_out;

    ohem_partial_kernel<<<NPARTIAL, 256, 0, stream>>>(preds, targets, partial);
    ohem_final_kernel<<<1, 64, 0, stream>>>(partial, out);
}